// AttentiveTransformer_5944234737744
// MI455X (gfx1250) — compile-verified
//
#include <hip/hip_runtime.h>
#include <hip/hip_bf16.h>
#include <math.h>

// ---------------------------------------------------------------------------
// Problem constants (from reference)
// ---------------------------------------------------------------------------
#define BS      640
#define IN_CH   2048
#define NSEQ    174          // S
#define EDIM    4            // E
#define OUT_CH  696          // S*E
#define CHUNK_R 320          // BS / 2 ghost-BN chunk rows
#define EPS     1e-5f

// GEMM tiling: block = 64(M) x 128(N), 8 waves in 2x4 grid, 32x32 per wave
#define BLK_M 64
#define BLK_N 128
#define BLK_K 32
#define A_STR 36             // floats/row: 144B -> 16B aligned rows for async B128
#define B_STR 136            // floats/row: +8 bank rotation per k-row

typedef __attribute__((ext_vector_type(2))) float v2f;
typedef __attribute__((ext_vector_type(8))) float v8f;
typedef __attribute__((ext_vector_type(4))) int   v4i;

// Optional gfx1250 async global->LDS path (ASYNCcnt-tracked), with fallback.
#if defined(__has_builtin)
#if __has_builtin(__builtin_amdgcn_global_load_async_to_lds_b128) && \
    __has_builtin(__builtin_amdgcn_s_wait_asynccnt)
#define HAVE_ASYNC_LDS 1
#endif
#endif
#ifndef HAVE_ASYNC_LDS
#define HAVE_ASYNC_LDS 0
#endif

// ---------------------------------------------------------------------------
// Kernel 1: H = X (640x2048) @ W^T (2048x696) via V_WMMA_F32_16X16X4_F32
// Each wave computes a 32x32 tile as 2x2 accumulators (4 independent chains).
// ---------------------------------------------------------------------------
__global__ __launch_bounds__(256)
void at_gemm_wmma_f32(const float* __restrict__ X,
                      const float* __restrict__ W,
                      float* __restrict__ H)
{
    __shared__ __align__(16) float lA[BLK_M * A_STR];   // [m][k]
    __shared__ __align__(16) float lB[BLK_K * B_STR];   // [k][n]  (transposed W)

    const int t    = threadIdx.x;
    const int lane = t & 31;
    const int wv   = t >> 5;        // 0..7
    const int wm   = wv >> 2;       // 0..1 -> M offset wm*32
    const int wn   = wv & 3;        // 0..3 -> N offset wn*32
    const int m0   = blockIdx.x * BLK_M;
    const int n0   = blockIdx.y * BLK_N;

    v8f acc[2][2] = {};

    // cooperative global->LDS indices
    const int arow = t >> 2;          // A row 0..63
    const int acol = (t & 3) << 3;    // A k offset 0,8,16,24
    const int brow = t >> 1;          // W row (= output channel) 0..127
    const int bkc  = (t & 1) << 4;    // k offset 0 or 16

    const int hi2 = (lane >> 4) << 1;   // 0 for lanes 0-15, 2 for lanes 16-31
    const int hi  = lane >> 4;
    const int ln  = lane & 15;
    const bool bvalid = (n0 + brow) < OUT_CH;

    for (int k0 = 0; k0 < IN_CH; k0 += BLK_K) {
        // --- stage A tile: 8 floats/thread, row-major copy ---
        {
            const float* gsrc = &X[(size_t)(m0 + arow) * IN_CH + k0 + acol];
            float* ldst = &lA[arow * A_STR + acol];
#if HAVE_ASYNC_LDS
            __builtin_amdgcn_global_load_async_to_lds_b128(
                (v4i*)gsrc, (v4i*)ldst, 0, 0);
            __builtin_amdgcn_global_load_async_to_lds_b128(
                (v4i*)(gsrc + 4), (v4i*)(ldst + 4), 0, 0);
#else
            const float4* gp = reinterpret_cast<const float4*>(gsrc);
            float4* lp = reinterpret_cast<float4*>(ldst);
            lp[0] = gp[0];
            lp[1] = gp[1];
#endif
        }
        // --- stage B tile transposed: lB[k][n] = W[n0+n][k0+k], 16 floats/thread ---
        if (bvalid) {
            const float4* wp = reinterpret_cast<const float4*>(
                &W[(size_t)(n0 + brow) * IN_CH + k0 + bkc]);
            #pragma unroll
            for (int q = 0; q < 4; ++q) {
                const float4 b4 = wp[q];
                lB[(bkc + q * 4 + 0) * B_STR + brow] = b4.x;
                lB[(bkc + q * 4 + 1) * B_STR + brow] = b4.y;
                lB[(bkc + q * 4 + 2) * B_STR + brow] = b4.z;
                lB[(bkc + q * 4 + 3) * B_STR + brow] = b4.w;
            }
        } else {
            #pragma unroll
            for (int j = 0; j < 16; ++j) lB[(bkc + j) * B_STR + brow] = 0.f;
        }
        // prefetch next K stage while this one is consumed
        if (k0 + BLK_K < IN_CH) {
            __builtin_prefetch(&X[(size_t)(m0 + arow) * IN_CH + k0 + BLK_K + acol], 0, 0);
            if (bvalid)
                __builtin_prefetch(&W[(size_t)(n0 + brow) * IN_CH + k0 + BLK_K + bkc], 0, 0);
        }
#if HAVE_ASYNC_LDS
        __builtin_amdgcn_s_wait_asynccnt(0);
#endif
        __syncthreads();

        // --- 8 K-steps of 4, 2x2 tiles -> 32 WMMA per stage ---
        #pragma unroll
        for (int kk = 0; kk < BLK_K; kk += 4) {
            v2f a[2], b[2];
            #pragma unroll
            for (int ti = 0; ti < 2; ++ti)
                a[ti] = *reinterpret_cast<const v2f*>(
                    &lA[(wm * 32 + ti * 16 + ln) * A_STR + kk + hi2]);
            #pragma unroll
            for (int tj = 0; tj < 2; ++tj) {
                b[tj].x = lB[(kk + hi2 + 0) * B_STR + wn * 32 + tj * 16 + ln];
                b[tj].y = lB[(kk + hi2 + 1) * B_STR + wn * 32 + tj * 16 + ln];
            }
            #pragma unroll
            for (int ti = 0; ti < 2; ++ti)
                #pragma unroll
                for (int tj = 0; tj < 2; ++tj)
                    acc[ti][tj] = __builtin_amdgcn_wmma_f32_16x16x4_f32(
                        false, a[ti], false, b[tj], (short)0, acc[ti][tj],
                        false, false);
        }
        __syncthreads();
    }

    // --- store: VGPR i of each tile holds rows i (lanes 0-15) / i+8 (16-31) ---
    #pragma unroll
    for (int ti = 0; ti < 2; ++ti) {
        #pragma unroll
        for (int tj = 0; tj < 2; ++tj) {
            const int col = n0 + wn * 32 + tj * 16 + ln;
            if (col < OUT_CH) {
                #pragma unroll
                for (int i = 0; i < 8; ++i) {
                    const int row = m0 + wm * 32 + ti * 16 + i + 8 * hi;
                    H[(size_t)row * OUT_CH + col] = acc[ti][tj][i];
                }
            }
        }
    }
}

// ---------------------------------------------------------------------------
// Kernel 2: ghost-BN statistics -> per (chunk, channel) scale/shift
// ---------------------------------------------------------------------------
__global__ void at_bn_stats(const float* __restrict__ H,
                            const float* __restrict__ gamma,
                            const float* __restrict__ beta,
                            float* __restrict__ scale,
                            float* __restrict__ shift)
{
    const int idx = blockIdx.x * blockDim.x + threadIdx.x;
    if (idx >= 2 * OUT_CH) return;
    const int chunk = idx / OUT_CH;
    const int ch    = idx - chunk * OUT_CH;
    const float* p  = H + (size_t)chunk * CHUNK_R * OUT_CH + ch;
    float s = 0.f, s2 = 0.f;
    for (int r = 0; r < CHUNK_R; ++r) {
        const float v = p[(size_t)r * OUT_CH];
        s  += v;
        s2 += v * v;
    }
    const float mean = s * (1.f / CHUNK_R);
    const float var  = s2 * (1.f / CHUNK_R) - mean * mean;
    const float sc   = gamma[ch] * rsqrtf(var + EPS);
    scale[idx] = sc;
    shift[idx] = beta[ch] - mean * sc;
}

// ---------------------------------------------------------------------------
// Kernel 3: apply BN * prior (in place into H) and compute Q/K/V (E=4 proj)
// ---------------------------------------------------------------------------
__global__ void at_apply_qkv(float* __restrict__ H,
                             const float* __restrict__ prior,
                             const float* __restrict__ inw,   // (12,4) row-major
                             const float* __restrict__ inb,   // (12,)
                             const float* __restrict__ scale,
                             const float* __restrict__ shift,
                             float* __restrict__ Q,
                             float* __restrict__ K,
                             float* __restrict__ V)
{
    const int idx = blockIdx.x * blockDim.x + threadIdx.x;
    if (idx >= BS * NSEQ) return;
    const int l = idx / NSEQ;
    const int n = idx - l * NSEQ;
    const int c0 = n << 2;
    const int chunk = (l >= CHUNK_R) ? 1 : 0;
    const size_t base = (size_t)l * OUT_CH + c0;

    float hb[4], pr[4];
    #pragma unroll
    for (int j = 0; j < 4; ++j) {
        const float sc = scale[chunk * OUT_CH + c0 + j];
        const float sh = shift[chunk * OUT_CH + c0 + j];
        pr[j] = prior[base + j];
        hb[j] = pr[j] * (H[base + j] * sc + sh);
        H[base + j] = hb[j];
    }
    #pragma unroll
    for (int e = 0; e < 4; ++e) {
        float q = inb[e], k = inb[4 + e], v = inb[8 + e];
        #pragma unroll
        for (int j = 0; j < 4; ++j) {
            q += inw[e * 4 + j]       * hb[j];
            k += inw[(4 + e) * 4 + j] * pr[j];
            v += inw[(8 + e) * 4 + j] * pr[j];
        }
        Q[base + e] = q;
        K[base + e] = k;
        V[base + e] = v;
    }
}

// ---------------------------------------------------------------------------
// Kernel 4: rank-1 attention per channel c
// ---------------------------------------------------------------------------
__global__ __launch_bounds__(256)
void at_attention(const float* __restrict__ Q,
                  const float* __restrict__ K,
                  const float* __restrict__ V,
                  float* __restrict__ O)
{
    __shared__ float lk[BS];
    __shared__ float lv[BS];
    __shared__ float red[256];

    const int c = blockIdx.x;
    const int t = threadIdx.x;

    float kmx = -3.4e38f, kmn = 3.4e38f;
    for (int s = t; s < BS; s += 256) {
        const float kv = K[(size_t)s * OUT_CH + c];
        lk[s] = kv;
        lv[s] = V[(size_t)s * OUT_CH + c];
        kmx = fmaxf(kmx, kv);
        kmn = fminf(kmn, kv);
    }
    red[t] = kmx; __syncthreads();
    for (int off = 128; off > 0; off >>= 1) {
        if (t < off) red[t] = fmaxf(red[t], red[t + off]);
        __syncthreads();
    }
    kmx = red[0]; __syncthreads();
    red[t] = kmn; __syncthreads();
    for (int off = 128; off > 0; off >>= 1) {
        if (t < off) red[t] = fminf(red[t], red[t + off]);
        __syncthreads();
    }
    kmn = red[0]; __syncthreads();

    for (int l = t; l < BS; l += 256) {
        const float q = Q[(size_t)l * OUT_CH + c];
        const float m = (q >= 0.f) ? q * kmx : q * kmn;   // max_s q*k[s]
        float num = 0.f, den = 0.f;
        for (int s = 0; s < BS; ++s) {
            const float ex = __expf(q * lk[s] - m);
            den += ex;
            num += ex * lv[s];
        }
        O[(size_t)l * OUT_CH + c] = num / den;
    }
}

// ---------------------------------------------------------------------------
// Kernel 5: out-proj + residual + row softmax over 696 channels
// ---------------------------------------------------------------------------
__global__ __launch_bounds__(256)
void at_finalize(const float* __restrict__ H,
                 const float* __restrict__ O,
                 const float* __restrict__ ow,   // (4,4)
                 const float* __restrict__ ob,   // (4,)
                 float* __restrict__ out)
{
    __shared__ float red[256];
    const int l = blockIdx.x;
    const int t = threadIdx.x;

    float hf[3];
    float mx = -3.4e38f;
    #pragma unroll
    for (int i = 0; i < 3; ++i) {
        const int c = t + i * 256;
        float val = -3.4e38f;
        if (c < OUT_CH) {
            const int n = c >> 2, e = c & 3;
            const size_t b = (size_t)l * OUT_CH + (n << 2);
            float y = ob[e];
            #pragma unroll
            for (int j = 0; j < 4; ++j) y += ow[e * 4 + j] * O[b + j];
            val = H[(size_t)l * OUT_CH + c] + y;
        }
        hf[i] = val;
        mx = fmaxf(mx, val);
    }
    red[t] = mx; __syncthreads();
    for (int off = 128; off > 0; off >>= 1) {
        if (t < off) red[t] = fmaxf(red[t], red[t + off]);
        __syncthreads();
    }
    mx = red[0]; __syncthreads();

    float ex[3];
    float se = 0.f;
    #pragma unroll
    for (int i = 0; i < 3; ++i) {
        const int c = t + i * 256;
        ex[i] = 0.f;
        if (c < OUT_CH) {
            ex[i] = __expf(hf[i] - mx);
            se += ex[i];
        }
    }
    red[t] = se; __syncthreads();
    for (int off = 128; off > 0; off >>= 1) {
        if (t < off) red[t] += red[t + off];
        __syncthreads();
    }
    const float inv = 1.f / red[0];
    #pragma unroll
    for (int i = 0; i < 3; ++i) {
        const int c = t + i * 256;
        if (c < OUT_CH) out[(size_t)l * OUT_CH + c] = ex[i] * inv;
    }
}

// ---------------------------------------------------------------------------
// Host launcher
// ---------------------------------------------------------------------------
extern "C" void kernel_launch(void* const* d_in, const int* in_sizes, int n_in,
                              void* d_out, int out_size, void* d_ws, size_t ws_size,
                              hipStream_t stream)
{
    (void)in_sizes; (void)n_in; (void)out_size; (void)ws_size;
    const float* x     = (const float*)d_in[0];
    const float* prior = (const float*)d_in[1];
    const float* w_lin = (const float*)d_in[2];
    const float* gamma = (const float*)d_in[3];
    const float* beta  = (const float*)d_in[4];
    const float* inw   = (const float*)d_in[5];
    const float* inb   = (const float*)d_in[6];
    const float* ow    = (const float*)d_in[7];
    const float* ob    = (const float*)d_in[8];

    float* ws  = (float*)d_ws;
    float* H   = ws;
    float* Q   = H + (size_t)BS * OUT_CH;
    float* K   = Q + (size_t)BS * OUT_CH;
    float* V   = K + (size_t)BS * OUT_CH;
    float* O   = V + (size_t)BS * OUT_CH;
    float* scl = O + (size_t)BS * OUT_CH;
    float* shf = scl + 2 * OUT_CH;

    dim3 ggrid(BS / BLK_M, (OUT_CH + BLK_N - 1) / BLK_N);   // 10 x 6
    at_gemm_wmma_f32<<<ggrid, 256, 0, stream>>>(x, w_lin, H);

    at_bn_stats<<<(2 * OUT_CH + 255) / 256, 256, 0, stream>>>(H, gamma, beta, scl, shf);

    at_apply_qkv<<<(BS * NSEQ + 255) / 256, 256, 0, stream>>>(
        H, prior, inw, inb, scl, shf, Q, K, V);

    at_attention<<<OUT_CH, 256, 0, stream>>>(Q, K, V, O);

    at_finalize<<<BS, 256, 0, stream>>>(H, O, ow, ob, (float*)d_out);
}